// EdgeWeightedGCNConv_17763984736712
// MI455X (gfx1250) — compile-verified
//
#include <hip/hip_runtime.h>
#include <math.h>

typedef __attribute__((ext_vector_type(2))) float v2f;
typedef __attribute__((ext_vector_type(8))) float v8f;

#define N_NODES 50000
#define N_EDGES 800000
#define HID 256

// ---------------------------------------------------------------------------
// Stage 1: per-edge scalar weight  w = sigmoid(W2 . silu(attr @ W1 + b1) + b2)
// ---------------------------------------------------------------------------
__global__ __launch_bounds__(256) void edge_mlp_kernel(
    const float* __restrict__ attr,  // [E,3]
    const float* __restrict__ W1,    // [3,128]
    const float* __restrict__ b1,    // [128]
    const float* __restrict__ W2,    // [128]
    const float* __restrict__ b2,    // [1]
    float* __restrict__ w, int E) {
  __shared__ float sW1[3 * 128];
  __shared__ float sb1[128];
  __shared__ float sW2[128];
  for (int i = threadIdx.x; i < 384; i += 256) sW1[i] = W1[i];
  if (threadIdx.x < 128) {
    sb1[threadIdx.x] = b1[threadIdx.x];
    sW2[threadIdx.x] = W2[threadIdx.x];
  }
  __syncthreads();
  int e = blockIdx.x * 256 + threadIdx.x;
  if (e >= E) return;
  float a0 = attr[e * 3 + 0];
  float a1 = attr[e * 3 + 1];
  float a2 = attr[e * 3 + 2];
  float acc = 0.0f;
#pragma unroll 4
  for (int j = 0; j < 128; ++j) {
    float z = fmaf(a0, sW1[j], fmaf(a1, sW1[128 + j], fmaf(a2, sW1[256 + j], sb1[j])));
    float t = z / (1.0f + __expf(-z));  // SiLU
    acc = fmaf(t, sW2[j], acc);
  }
  float zz = acc + b2[0];
  w[e] = 1.0f / (1.0f + __expf(-zz));  // sigmoid
}

// ---------------------------------------------------------------------------
// Stage 2: degree = 1.0 (self loop) + scatter(w at col); then dinv = rsqrt(deg)
// ---------------------------------------------------------------------------
__global__ __launch_bounds__(256) void deg_init_kernel(float* __restrict__ deg, int N) {
  int i = blockIdx.x * 256 + threadIdx.x;
  if (i < N) deg[i] = 1.0f;  // self-loop weight
}

__global__ __launch_bounds__(256) void deg_scatter_kernel(
    const int* __restrict__ col, const float* __restrict__ w,
    float* __restrict__ deg, int E) {
  int e = blockIdx.x * 256 + threadIdx.x;
  if (e < E) unsafeAtomicAdd(&deg[col[e]], w[e]);
}

__global__ __launch_bounds__(256) void deg_rsqrt_kernel(float* __restrict__ deg, int N) {
  int i = blockIdx.x * 256 + threadIdx.x;
  if (i < N) {
    float d = deg[i];
    deg[i] = d > 0.0f ? rsqrtf(d) : 0.0f;
  }
}

// ---------------------------------------------------------------------------
// Stage 3: x = h @ Wc via V_WMMA_F32_16X16X4_F32 (exact fp32, matrix pipe).
// M=50000 (3125 tiles), N=256 (16 tiles), K=256. One wave -> one 16x16 tile.
// Block = 8 waves sharing the same 16 rows of h (different N tiles).
// Grid = 3125 * 2 blocks (blockIdx & 1 selects N-half).
// ---------------------------------------------------------------------------
__global__ __launch_bounds__(256) void gemm_xWc_kernel(
    const float* __restrict__ h,   // [M,256]
    const float* __restrict__ Wc,  // [256,256]
    float* __restrict__ x) {       // [M,256]
  const int K = 256, NN = 256;
  const int wave = threadIdx.x >> 5;
  const int lane = threadIdx.x & 31;
  const int m0 = (blockIdx.x >> 1) * 16;
  const int n0 = (((blockIdx.x & 1) << 3) + wave) * 16;
  const int r = lane & 15;   // M index within A-frag / N index within B,C,D
  const int g = lane >> 4;   // K pair selector (0 -> K{0,1}, 1 -> K{2,3})

  const float* arow = h + (size_t)(m0 + r) * K + 2 * g;       // A[m0+r][2g + kk]
  const float* bcol = Wc + (size_t)(2 * g) * NN + n0 + r;     // B[2g + kk][n0+r]

  v8f c = {};
  for (int kk = 0; kk < K; kk += 4) {
    v2f a;
    a.x = arow[kk];
    a.y = arow[kk + 1];
    v2f b;
    b.x = bcol[kk * NN];
    b.y = bcol[kk * NN + NN];
    // (neg_a, A, neg_b, B, c_mod, C, reuse_a, reuse_b)
    c = __builtin_amdgcn_wmma_f32_16x16x4_f32(false, a, false, b, (short)0, c,
                                              false, false);
  }
#pragma unroll
  for (int i = 0; i < 8; ++i) {
    int m = m0 + i + 8 * g;      // C/D layout: VGPR i, lanes 16-31 -> M=i+8
    x[(size_t)m * NN + n0 + r] = c[i];
  }
}

// ---------------------------------------------------------------------------
// Stage 4: self-loop init  out[n][j] = x[n][j]*dinv[n]^2 + bc[j]
// ---------------------------------------------------------------------------
__global__ __launch_bounds__(256) void out_init_kernel(
    const float* __restrict__ x, const float* __restrict__ dinv,
    const float* __restrict__ bc, float* __restrict__ out) {
  int idx = blockIdx.x * 256 + threadIdx.x;  // grid = N_NODES blocks
  int n = idx >> 8;
  int j = idx & 255;
  float di = dinv[n];
  out[idx] = x[idx] * di * di + bc[j];
}

// ---------------------------------------------------------------------------
// Stage 5: gather-scale-scatter. One block per edge; row/col/w are uniform
// (scalarized loads); 256 lanes cover HID with hardware fp32 atomics.
// ---------------------------------------------------------------------------
__global__ __launch_bounds__(256) void aggregate_kernel(
    const int* __restrict__ ei,    // [2,E] flat: rows then cols
    const float* __restrict__ w,   // [E]
    const float* __restrict__ dinv,
    const float* __restrict__ x,   // [N,256]
    float* __restrict__ out, int E) {
  int e = blockIdx.x;
  int r = ei[e];
  int c = ei[E + e];
  float norm = dinv[r] * w[e] * dinv[c];
  int j = threadIdx.x;
  float v = x[(size_t)r * HID + j] * norm;
  unsafeAtomicAdd(&out[(size_t)c * HID + j], v);
}

// ---------------------------------------------------------------------------
extern "C" void kernel_launch(void* const* d_in, const int* in_sizes, int n_in,
                              void* d_out, int out_size, void* d_ws, size_t ws_size,
                              hipStream_t stream) {
  const float* h    = (const float*)d_in[0];
  const int*   ei   = (const int*)d_in[1];   // [2,E]
  const float* attr = (const float*)d_in[2];
  const float* W1   = (const float*)d_in[3];
  const float* b1   = (const float*)d_in[4];
  const float* W2   = (const float*)d_in[5];
  const float* b2   = (const float*)d_in[6];
  const float* Wc   = (const float*)d_in[7];
  const float* bc   = (const float*)d_in[8];
  float* out = (float*)d_out;

  const int N = N_NODES, E = N_EDGES;

  // workspace layout: x [N*HID] | w [E] | deg/dinv [N]   (~54.6 MB)
  float* x   = (float*)d_ws;
  float* w   = x + (size_t)N * HID;
  float* deg = w + E;

  // 1: edge weights
  edge_mlp_kernel<<<(E + 255) / 256, 256, 0, stream>>>(attr, W1, b1, W2, b2, w, E);
  // 2: degrees -> dinv (in place)
  deg_init_kernel<<<(N + 255) / 256, 256, 0, stream>>>(deg, N);
  deg_scatter_kernel<<<(E + 255) / 256, 256, 0, stream>>>(ei + E, w, deg, E);
  deg_rsqrt_kernel<<<(N + 255) / 256, 256, 0, stream>>>(deg, N);
  // 3: x = h @ Wc  (WMMA fp32)
  gemm_xWc_kernel<<<(N / 16) * 2, 256, 0, stream>>>(h, Wc, x);
  // 4: self loops + bias
  out_init_kernel<<<N, 256, 0, stream>>>(x, deg, bc, out);
  // 5: edge aggregation
  aggregate_kernel<<<E, 256, 0, stream>>>(ei, w, deg, x, out, E);
}